// Seq2SeqAttn_55525337203101
// MI455X (gfx1250) — compile-verified
//
#include <hip/hip_runtime.h>
#include <math.h>

typedef __attribute__((ext_vector_type(2))) float v2f;
typedef __attribute__((ext_vector_type(8))) float v8f;

#define B_   256
#define L_   128
#define E_   300
#define H_   300
#define O_   30
#define W_   5
#define GI3  (3 * H_)     // 900
#define HD_  (2 * H_)     // 600
#define GD3  (3 * HD_)    // 1800

// LDS strides (odd -> bank-conflict-free column walks)
#define HSTR  301
#define GSTR  901
#define HDSTR 601
#define GDSTR 1801

// ---------------------------------------------------------------------------
// Embedding gather: tf1[b,l,:] = emb[x[b,l],:]
// ---------------------------------------------------------------------------
__global__ void k_embed(const int* __restrict__ x, const float* __restrict__ emb,
                        float* __restrict__ tf1, int total) {
  int i = blockIdx.x * blockDim.x + threadIdx.x;
  int stride = gridDim.x * blockDim.x;
  for (; i < total; i += stride) {
    int row = i / E_;
    int col = i - row * E_;
    tf1[i] = emb[(long)x[row] * E_ + col];
  }
}

// ---------------------------------------------------------------------------
// G[m,n] = sum_k X[m,k] * Wt[n,k] + bias[n]
// M multiple of 128, K multiple of 4, N edge handled branchlessly via
// clamped row pointer + 0/1 mask (keeps the k-loop free of exec juggling).
// Block = 256 threads = 8 waves; wave w owns M-tile m0=blk.x*128+w*16,
// all waves share the same 16-col B strip (weight reuse in cache).
// One fp32 WMMA 16x16x4 per k-step.
// ---------------------------------------------------------------------------
__global__ void k_gemm_bias(const float* __restrict__ X, const float* __restrict__ Wt,
                            const float* __restrict__ bias, float* __restrict__ G,
                            int M, int N, int K) {
  const int lane = threadIdx.x & 31;
  const int wave = threadIdx.x >> 5;
  const int half = lane >> 4;     // selects K pair 0..1 vs 2..3
  const int r    = lane & 15;     // M index within A-tile / N index within B-tile
  const int m0   = blockIdx.x * 128 + wave * 16;
  const int n0   = blockIdx.y * 16;
  const int n    = n0 + r;
  const bool nok = (n < N);
  const float msk  = nok ? 1.f : 0.f;
  const int   ncl  = nok ? n : (N - 1);          // clamped valid row
  const float* xrow = X + (long)(m0 + r) * K + 2 * half;
  const float* wrow = Wt + (long)ncl * K + 2 * half;

  v8f acc = {};
  for (int k0 = 0; k0 < K; k0 += 64) {
    __builtin_prefetch(xrow + k0 + 64, 0, 1);    // -> global_prefetch_b8
    __builtin_prefetch(wrow + k0 + 64, 0, 1);
    const int kend = (k0 + 64 < K) ? (k0 + 64) : K;
    for (int k = k0; k < kend; k += 4) {
      v2f a, b;
      a.x = xrow[k];
      a.y = xrow[k + 1];
      b.x = wrow[k]     * msk;
      b.y = wrow[k + 1] * msk;
      acc = __builtin_amdgcn_wmma_f32_16x16x4_f32(false, a, false, b,
                                                  (short)0, acc, false, false);
    }
  }
  if (nok) {
    const float bv = bias[n];
#pragma unroll
    for (int j = 0; j < 8; ++j) {
      const int m = m0 + j + (half ? 8 : 0);     // C/D layout: VGPR j -> M=j / M=8+j
      G[(long)m * N + n] = acc[j] + bv;
    }
  }
}

// ---------------------------------------------------------------------------
// GRU time scan, layers 1 & 2. One workgroup (1024 thr = 32 waves) owns a
// 16-row batch tile for one time direction (blockIdx.y = dir) and loops all
// L timesteps with h resident in LDS. Per step:
//   phase1: gh = h @ whh^T + bhh   (57 WMMA column tiles across 32 waves)
//   phase2: elementwise GRU gate math, update h, emit y / final h.
// ---------------------------------------------------------------------------
__global__ void k_gru_scan(const float* __restrict__ giF, const float* __restrict__ giB,
                           const float* __restrict__ whhF, const float* __restrict__ bhhF,
                           const float* __restrict__ whhB, const float* __restrict__ bhhB,
                           float* __restrict__ y,      // [B][L][600] (layer1) or nullptr
                           float* __restrict__ hfin) { // [B][600]    (layer2) or nullptr
  extern __shared__ float smem[];
  float* hS  = smem;             // 16 x HSTR
  float* ghS = smem + 16 * HSTR; // 16 x GSTR

  const int dir = blockIdx.y;
  const float* gi  = dir ? giB  : giF;
  const float* whh = dir ? whhB : whhF;
  const float* bhh = dir ? bhhB : bhhF;
  const int b0   = blockIdx.x * 16;
  const int tid  = threadIdx.x;
  const int lane = tid & 31, wave = tid >> 5;
  const int half = lane >> 4, r = lane & 15;

  for (int i = tid; i < 16 * HSTR; i += blockDim.x) hS[i] = 0.f;
  __syncthreads();

  for (int s = 0; s < L_; ++s) {
    const int t = dir ? (L_ - 1 - s) : s;

    // phase 1: gh tiles
    for (int tile = wave; tile < 57; tile += 32) {
      const int n0 = tile * 16;
      const int n  = n0 + r;
      const bool nok = (n < GI3);
      const float msk = nok ? 1.f : 0.f;
      const int   ncl = nok ? n : (GI3 - 1);
      const float* wrow = whh + (long)ncl * H_ + 2 * half;
      const float* hrow = hS + r * HSTR + 2 * half;
      v8f acc = {};
      for (int k = 0; k < H_; k += 4) {
        v2f a, b;
        a.x = hrow[k];
        a.y = hrow[k + 1];
        b.x = wrow[k]     * msk;
        b.y = wrow[k + 1] * msk;
        acc = __builtin_amdgcn_wmma_f32_16x16x4_f32(false, a, false, b,
                                                    (short)0, acc, false, false);
      }
      if (nok) {
        const float bv = bhh[n];
#pragma unroll
        for (int j = 0; j < 8; ++j)
          ghS[(j + (half ? 8 : 0)) * GSTR + n] = acc[j] + bv;
      }
    }
    __syncthreads();

    // phase 2: gate math + h update
    for (int e = tid; e < 16 * H_; e += blockDim.x) {
      const int m = e / H_;
      const int j = e - m * H_;
      const int bb = b0 + m;
      const float* g = gi + ((long)bb * L_ + t) * GI3;
      const float ir = g[j], iz = g[H_ + j], inn = g[2 * H_ + j];
      const float hr = ghS[m * GSTR + j];
      const float hz = ghS[m * GSTR + H_ + j];
      const float hn = ghS[m * GSTR + 2 * H_ + j];
      const float rg = 1.f / (1.f + expf(-(ir + hr)));
      const float zg = 1.f / (1.f + expf(-(iz + hz)));
      const float ng = tanhf(inn + rg * hn);
      const float hnew = (1.f - zg) * ng + zg * hS[m * HSTR + j];
      hS[m * HSTR + j] = hnew;
      if (y) y[((long)bb * L_ + t) * HD_ + dir * H_ + j] = hnew;
    }
    __syncthreads();
  }

  if (hfin) {
    for (int e = tid; e < 16 * H_; e += blockDim.x) {
      const int m = e / H_, j = e - m * H_;
      hfin[(long)(b0 + m) * HD_ + dir * H_ + j] = hS[m * HSTR + j];
    }
  }
}

// ---------------------------------------------------------------------------
// Build the 256 sliding-window buffers from utt / forehead0.
// buf_w[j] = utt[w-4+j] if w-4+j >= 0 else forehead0[w+j+1]
// ---------------------------------------------------------------------------
__global__ void k_build_buf(const float* __restrict__ utt, const float* __restrict__ fh,
                            float* __restrict__ buf) {
  int i = blockIdx.x * blockDim.x + threadIdx.x;
  const int total = B_ * W_ * HD_;
  const int stride = gridDim.x * blockDim.x;
  for (; i < total; i += stride) {
    const int c = i % HD_;
    const int rest = i / HD_;
    const int j = rest % W_;
    const int w = rest / W_;
    const int src = w - (W_ - 1) + j;
    buf[i] = (src >= 0) ? utt[(long)src * HD_ + c]
                        : fh[(long)(w + j + 1) * HD_ + c];
  }
}

// ---------------------------------------------------------------------------
// Decoder GRU over the W=5 window rows, batched across all 256 independent
// windows. Same persistent-LDS pattern as k_gru_scan, H=600, gates=1800.
// ---------------------------------------------------------------------------
__global__ void k_dec_scan(const float* __restrict__ gi,   // [256][5][1800]
                           const float* __restrict__ whh, const float* __restrict__ bhh,
                           float* __restrict__ win) {      // [256][5][600]
  extern __shared__ float smem[];
  float* hS  = smem;              // 16 x HDSTR
  float* ghS = smem + 16 * HDSTR; // 16 x GDSTR

  const int b0   = blockIdx.x * 16;
  const int tid  = threadIdx.x;
  const int lane = tid & 31, wave = tid >> 5;
  const int half = lane >> 4, r = lane & 15;

  for (int i = tid; i < 16 * HDSTR; i += blockDim.x) hS[i] = 0.f;
  __syncthreads();

  for (int t = 0; t < W_; ++t) {
    for (int tile = wave; tile < 113; tile += 32) {
      const int n0 = tile * 16;
      const int n  = n0 + r;
      const bool nok = (n < GD3);
      const float msk = nok ? 1.f : 0.f;
      const int   ncl = nok ? n : (GD3 - 1);
      const float* wrow = whh + (long)ncl * HD_ + 2 * half;
      const float* hrow = hS + r * HDSTR + 2 * half;
      v8f acc = {};
      for (int k = 0; k < HD_; k += 4) {
        v2f a, b;
        a.x = hrow[k];
        a.y = hrow[k + 1];
        b.x = wrow[k]     * msk;
        b.y = wrow[k + 1] * msk;
        acc = __builtin_amdgcn_wmma_f32_16x16x4_f32(false, a, false, b,
                                                    (short)0, acc, false, false);
      }
      if (nok) {
        const float bv = bhh[n];
#pragma unroll
        for (int j = 0; j < 8; ++j)
          ghS[(j + (half ? 8 : 0)) * GDSTR + n] = acc[j] + bv;
      }
    }
    __syncthreads();

    for (int e = tid; e < 16 * HD_; e += blockDim.x) {
      const int m = e / HD_, j = e - m * HD_;
      const int w = b0 + m;
      const float* g = gi + ((long)w * W_ + t) * GD3;
      const float ir = g[j], iz = g[HD_ + j], inn = g[2 * HD_ + j];
      const float hr = ghS[m * GDSTR + j];
      const float hz = ghS[m * GDSTR + HD_ + j];
      const float hn = ghS[m * GDSTR + 2 * HD_ + j];
      const float rg = 1.f / (1.f + expf(-(ir + hr)));
      const float zg = 1.f / (1.f + expf(-(iz + hz)));
      const float ng = tanhf(inn + rg * hn);
      const float hnew = (1.f - zg) * ng + zg * hS[m * HDSTR + j];
      hS[m * HDSTR + j] = hnew;
      win[((long)w * W_ + t) * HD_ + j] = hnew;
    }
    __syncthreads();
  }
}

// ---------------------------------------------------------------------------
// Per-window attention (Q=K=buf, V=win), LeakyReLU(softmax), ctx = rows-sum,
// then fused FC -> out[w, 0:30]. One block per window.
// ---------------------------------------------------------------------------
__global__ void k_attn_fc(const float* __restrict__ buf, const float* __restrict__ win,
                          const float* __restrict__ fcw, const float* __restrict__ fcb,
                          float* __restrict__ out) {
  __shared__ float S[W_ * W_];
  __shared__ float colw[W_];
  __shared__ float ctx[HD_];
  const int w = blockIdx.x;
  const int tid = threadIdx.x;
  const float* bw = buf + (long)w * W_ * HD_;
  const float* wn = win + (long)w * W_ * HD_;
  const float scale = 0.040824829046386304f; // 1/sqrt(600)

  for (int p = tid; p < W_ * W_; p += blockDim.x) {
    const int i = p / W_, j = p % W_;
    float s = 0.f;
    for (int c = 0; c < HD_; ++c) s += bw[i * HD_ + c] * bw[j * HD_ + c];
    S[p] = s * scale;
  }
  __syncthreads();

  if (tid < W_) {
    float mx = -1e30f;
    for (int j = 0; j < W_; ++j) mx = fmaxf(mx, S[tid * W_ + j]);
    float e[W_], sum = 0.f;
    for (int j = 0; j < W_; ++j) { e[j] = expf(S[tid * W_ + j] - mx); sum += e[j]; }
    for (int j = 0; j < W_; ++j) {
      float a = e[j] / sum;
      a = (a >= 0.f) ? a : 0.1f * a;    // LeakyReLU(0.1) per reference
      S[tid * W_ + j] = a;
    }
  }
  __syncthreads();

  if (tid < W_) {
    float cs = 0.f;
    for (int i = 0; i < W_; ++i) cs += S[i * W_ + tid];
    colw[tid] = cs;
  }
  __syncthreads();

  for (int c = tid; c < HD_; c += blockDim.x) {
    float v = 0.f;
    for (int j = 0; j < W_; ++j) v += colw[j] * wn[j * HD_ + c];
    ctx[c] = v;
  }
  __syncthreads();

  for (int o = tid; o < O_; o += blockDim.x) {
    float v = fcb[o];
    const float* fr = fcw + (long)o * HD_;
    for (int c = 0; c < HD_; ++c) v += fr[c] * ctx[c];
    out[(long)w * O_ + o] = v;
  }
}

// ---------------------------------------------------------------------------
extern "C" void kernel_launch(void* const* d_in, const int* in_sizes, int n_in,
                              void* d_out, int out_size, void* d_ws, size_t ws_size,
                              hipStream_t stream) {
  (void)in_sizes; (void)n_in; (void)out_size; (void)ws_size;
  const int*   x        = (const int*)d_in[0];
  const float* emb      = (const float*)d_in[1];
  const float* e1f_wih  = (const float*)d_in[2];
  const float* e1f_whh  = (const float*)d_in[3];
  const float* e1f_bih  = (const float*)d_in[4];
  const float* e1f_bhh  = (const float*)d_in[5];
  const float* e1b_wih  = (const float*)d_in[6];
  const float* e1b_whh  = (const float*)d_in[7];
  const float* e1b_bih  = (const float*)d_in[8];
  const float* e1b_bhh  = (const float*)d_in[9];
  const float* e2f_wih  = (const float*)d_in[10];
  const float* e2f_whh  = (const float*)d_in[11];
  const float* e2f_bih  = (const float*)d_in[12];
  const float* e2f_bhh  = (const float*)d_in[13];
  const float* e2b_wih  = (const float*)d_in[14];
  const float* e2b_whh  = (const float*)d_in[15];
  const float* e2b_bih  = (const float*)d_in[16];
  const float* e2b_bhh  = (const float*)d_in[17];
  const float* dec_wih  = (const float*)d_in[18];
  const float* dec_whh  = (const float*)d_in[19];
  const float* dec_bih  = (const float*)d_in[20];
  const float* dec_bhh  = (const float*)d_in[21];
  const float* fc_w     = (const float*)d_in[22];
  const float* fc_b     = (const float*)d_in[23];
  const float* fh0      = (const float*)d_in[24];

  float* ws = (float*)d_ws;
  size_t off = 0;
  float* tf1  = ws + off; off += (size_t)B_ * L_ * E_;      // 9.83 M
  float* Ga   = ws + off; off += (size_t)B_ * L_ * GI3;     // 29.5 M (layer1 fwd / layer2 fwd)
  float* Gb   = ws + off; off += (size_t)B_ * L_ * GI3;     // 29.5 M (layer1 bwd / layer2 bwd)
  float* tf2  = ws + off; off += (size_t)B_ * L_ * HD_;     // 19.7 M
  float* utt  = ws + off; off += (size_t)B_ * HD_;
  float* buf  = ws + off; off += (size_t)B_ * W_ * HD_;
  float* gid  = ws + off; off += (size_t)B_ * W_ * GD3;
  float* winb = ws + off; off += (size_t)B_ * W_ * HD_;

  // 1) embedding gather
  k_embed<<<4096, 256, 0, stream>>>(x, emb, tf1, B_ * L_ * E_);

  // 2) layer-1 input-side gate GEMMs (fwd/bwd)
  dim3 g1(B_ * L_ / 128, (GI3 + 15) / 16);
  k_gemm_bias<<<g1, 256, 0, stream>>>(tf1, e1f_wih, e1f_bih, Ga, B_ * L_, GI3, E_);
  k_gemm_bias<<<g1, 256, 0, stream>>>(tf1, e1b_wih, e1b_bih, Gb, B_ * L_, GI3, E_);

  // 3) layer-1 bidirectional scan -> tf2
  const size_t sh1 = (size_t)(16 * HSTR + 16 * GSTR) * sizeof(float);
  dim3 gs(B_ / 16, 2);
  k_gru_scan<<<gs, 1024, sh1, stream>>>(Ga, Gb, e1f_whh, e1f_bhh, e1b_whh, e1b_bhh,
                                        tf2, nullptr);

  // 4) layer-2 input-side gate GEMMs (reuse Ga/Gb)
  k_gemm_bias<<<g1, 256, 0, stream>>>(tf2, e2f_wih, e2f_bih, Ga, B_ * L_, GI3, HD_);
  k_gemm_bias<<<g1, 256, 0, stream>>>(tf2, e2b_wih, e2b_bih, Gb, B_ * L_, GI3, HD_);

  // 5) layer-2 bidirectional scan -> utt (final hiddens only)
  k_gru_scan<<<gs, 1024, sh1, stream>>>(Ga, Gb, e2f_whh, e2f_bhh, e2b_whh, e2b_bhh,
                                        nullptr, utt);

  // 6) materialize all 256 independent sliding windows
  k_build_buf<<<256, 256, 0, stream>>>(utt, fh0, buf);

  // 7) decoder input-side gate GEMM: [1280 x 600] x [600 x 1800]
  dim3 gd(B_ * W_ / 128, (GD3 + 15) / 16);
  k_gemm_bias<<<gd, 256, 0, stream>>>(buf, dec_wih, dec_bih, gid, B_ * W_, GD3, HD_);

  // 8) batched decoder GRU over W=5 window rows -> win
  const size_t sh2 = (size_t)(16 * HDSTR + 16 * GDSTR) * sizeof(float);
  k_dec_scan<<<B_ / 16, 1024, sh2, stream>>>(gid, dec_whh, dec_bhh, winb);

  // 9) per-window attention + fused FC -> d_out [256 x 30]
  k_attn_fc<<<B_, 128, 0, stream>>>(buf, winb, fc_w, fc_b, (float*)d_out);
}